// Proto_70196945485969
// MI455X (gfx1250) — compile-verified
//
#include <hip/hip_runtime.h>
#include <hip/hip_bf16.h>
#include <math.h>

// Problem constants (from reference setup_inputs)
#define BB 8
#define NN 20
#define KK 5
#define DD 230
#define NQ 200

typedef __attribute__((ext_vector_type(2))) float v2f;
typedef __attribute__((ext_vector_type(8))) float v8f;

__device__ __forceinline__ float dev_tanh(float x) {
#if __has_builtin(__builtin_amdgcn_tanhf)
    return __builtin_amdgcn_tanhf(x);
#else
    return tanhf(x);
#endif
}

// -------------------------------------------------------------------------
// Y[M x 230] = X[M x 230] @ W^T + bias, W is 230x230 row-major ([dout][din]).
// One wave computes a 16x64 strip of Y with fp32 WMMA V_WMMA_F32_16X16X4_F32:
// 4 accumulators share one A fragment per k-step (4 wmma : 5 b64 loads).
//
// fp32 A 16x4 layout (ISA 7.12.2): lane holds row M=lane&15; VGPR v holds
// K = k0 + v + 2*(lane>>4).  B 4x16: lane holds col N=lane&15, same K rule.
// C/D 16x16: lane holds col N=lane&15; VGPR v holds row M = v + 8*(lane>>4).
//
// M is a multiple of 16 (1600 / 800) -> no row guards. k tail (228..231):
// zero the A hi-half (K=230,231) via a lane-uniform multiply; A[:,k]=0 kills
// those k contributions regardless of B. Column overrun past 230: clamp the
// B row pointer (garbage only lands in D columns we never store).
// -------------------------------------------------------------------------
__global__ void __launch_bounds__(32)
fc_wmma_kernel(const float* __restrict__ X,
               const float* __restrict__ W,
               const float* __restrict__ bias,
               float* __restrict__ Y,
               int M) {
    const int D    = DD;
    const int lane = threadIdx.x;      // block = 32 threads = 1 wave
    const int lo   = lane & 15;
    const int hi   = lane >> 4;
    const int m0   = blockIdx.x * 16;
    const int n0   = blockIdx.y * 64;

    const float* Xr = X + (size_t)(m0 + lo) * D;

    const int nc0 = n0 + 0  + lo;
    const int nc1 = n0 + 16 + lo;
    const int nc2 = n0 + 32 + lo;
    const int nc3 = n0 + 48 + lo;
    const float* W0 = W + (size_t)(nc0 < D ? nc0 : D - 1) * D;
    const float* W1 = W + (size_t)(nc1 < D ? nc1 : D - 1) * D;
    const float* W2 = W + (size_t)(nc2 < D ? nc2 : D - 1) * D;
    const float* W3 = W + (size_t)(nc3 < D ? nc3 : D - 1) * D;

    v8f c0 = {}, c1 = {}, c2 = {}, c3 = {};

    // 57 full k-steps: k = 0..227, all in range, branch-free b64 loads
    for (int k0 = 0; k0 < 228; k0 += 4) {
        const int ka = k0 + 2 * hi;
        v2f a  = *(const v2f*)(Xr + ka);
        v2f b0 = *(const v2f*)(W0 + ka);
        v2f b1 = *(const v2f*)(W1 + ka);
        v2f b2 = *(const v2f*)(W2 + ka);
        v2f b3 = *(const v2f*)(W3 + ka);
        c0 = __builtin_amdgcn_wmma_f32_16x16x4_f32(false, a, false, b0, (short)0, c0, false, false);
        c1 = __builtin_amdgcn_wmma_f32_16x16x4_f32(false, a, false, b1, (short)0, c1, false, false);
        c2 = __builtin_amdgcn_wmma_f32_16x16x4_f32(false, a, false, b2, (short)0, c2, false, false);
        c3 = __builtin_amdgcn_wmma_f32_16x16x4_f32(false, a, false, b3, (short)0, c3, false, false);
    }

    // tail k-step: k = 228,229 valid; 230,231 zeroed through A
    {
        const float z = hi ? 0.f : 1.f;   // lane-uniform scale, no branches
        v2f a  = *(const v2f*)(Xr + 228);
        a.x *= z; a.y *= z;
        v2f b0 = *(const v2f*)(W0 + 228);
        v2f b1 = *(const v2f*)(W1 + 228);
        v2f b2 = *(const v2f*)(W2 + 228);
        v2f b3 = *(const v2f*)(W3 + 228);
        c0 = __builtin_amdgcn_wmma_f32_16x16x4_f32(false, a, false, b0, (short)0, c0, false, false);
        c1 = __builtin_amdgcn_wmma_f32_16x16x4_f32(false, a, false, b1, (short)0, c1, false, false);
        c2 = __builtin_amdgcn_wmma_f32_16x16x4_f32(false, a, false, b2, (short)0, c2, false, false);
        c3 = __builtin_amdgcn_wmma_f32_16x16x4_f32(false, a, false, b3, (short)0, c3, false, false);
    }

    const float bv0 = bias[nc0 < D ? nc0 : D - 1];
    const float bv1 = bias[nc1 < D ? nc1 : D - 1];
    const float bv2 = bias[nc2 < D ? nc2 : D - 1];
    const float bv3 = bias[nc3 < D ? nc3 : D - 1];

#pragma unroll
    for (int v = 0; v < 8; ++v) {
        const int row = m0 + v + 8 * hi;      // always < M
        float* Yr = Y + (size_t)row * D;
        if (nc0 < D) Yr[nc0] = c0[v] + bv0;
        if (nc1 < D) Yr[nc1] = c1[v] + bv1;
        if (nc2 < D) Yr[nc2] = c2[v] + bv2;
        if (nc3 < D) Yr[nc3] = c3[v] + bv3;
    }
}

// -------------------------------------------------------------------------
// Fused conv score: per (b,n) image (block), per D-column (thread):
//   x1[i][h] = relu(b1 + sum_dk w1[i,dk]*Spad[h+dk-2])         (1 -> 32 ch)
//   x2[o][h] = relu(b2 + sum_{i,dk} w2[o,i,dk]*x1pad[i,h+dk-2]) (32 -> 64)
//   score    = relu(bf + sum_{o,h} wf[o,h]*x2[o][h])            (stride-5 final)
// Scatter form h2 = h - dk + 2 keeps all x1 indices compile-time constant
// (x1 stays in VGPRs). Weights staged in LDS (~43 KB).
// -------------------------------------------------------------------------
__global__ void __launch_bounds__(256)
score_kernel(const float* __restrict__ S,
             const float* __restrict__ w1, const float* __restrict__ b1,
             const float* __restrict__ w2, const float* __restrict__ b2,
             const float* __restrict__ wf, const float* __restrict__ bf,
             float* __restrict__ score) {
    __shared__ float sw1[32 * 5];
    __shared__ float sb1[32];
    __shared__ float sw2[64 * 32 * 5];
    __shared__ float sb2[64];
    __shared__ float swf[64 * 5];
    __shared__ float sbf;

    const int tid = threadIdx.x;
    for (int i = tid; i < 32 * 5;      i += blockDim.x) sw1[i] = w1[i];
    for (int i = tid; i < 32;          i += blockDim.x) sb1[i] = b1[i];
    for (int i = tid; i < 64 * 32 * 5; i += blockDim.x) sw2[i] = w2[i];
    for (int i = tid; i < 64;          i += blockDim.x) sb2[i] = b2[i];
    for (int i = tid; i < 64 * 5;      i += blockDim.x) swf[i] = wf[i];
    if (tid == 0) sbf = bf[0];
    __syncthreads();

    const int img = blockIdx.x;     // 0 .. B*N-1
    const int d   = tid;
    if (d >= DD) return;

    const float* Sc = S + (size_t)img * KK * DD + d;
    float sv[KK];
#pragma unroll
    for (int k = 0; k < KK; ++k) sv[k] = Sc[k * DD];

    // conv1 -> x1[32][5] in registers
    float x1[32][5];
#pragma unroll
    for (int i = 0; i < 32; ++i) {
        float a0 = sb1[i], a1 = a0, a2 = a0, a3 = a0, a4 = a0;
#pragma unroll
        for (int h = 0; h < 5; ++h) {
#pragma unroll
            for (int dk = 0; dk < 5; ++dk) {
                const int h2 = h - dk + 2;
                if (h2 >= 0 && h2 < 5) {
                    const float p = sw1[i * 5 + dk] * sv[h];
                    if      (h2 == 0) a0 += p;
                    else if (h2 == 1) a1 += p;
                    else if (h2 == 2) a2 += p;
                    else if (h2 == 3) a3 += p;
                    else              a4 += p;
                }
            }
        }
        x1[i][0] = fmaxf(a0, 0.f); x1[i][1] = fmaxf(a1, 0.f);
        x1[i][2] = fmaxf(a2, 0.f); x1[i][3] = fmaxf(a3, 0.f);
        x1[i][4] = fmaxf(a4, 0.f);
    }

    // conv2 + convf fused: never materialize x2
    float out = sbf;
    for (int o = 0; o < 64; ++o) {
        float a0 = sb2[o], a1 = a0, a2 = a0, a3 = a0, a4 = a0;
        const float* w2o = &sw2[o * 32 * 5];
#pragma unroll
        for (int i = 0; i < 32; ++i) {
#pragma unroll
            for (int h = 0; h < 5; ++h) {
                const float xv = x1[i][h];
#pragma unroll
                for (int dk = 0; dk < 5; ++dk) {
                    const int h2 = h - dk + 2;
                    if (h2 >= 0 && h2 < 5) {
                        const float p = w2o[i * 5 + dk] * xv;
                        if      (h2 == 0) a0 += p;
                        else if (h2 == 1) a1 += p;
                        else if (h2 == 2) a2 += p;
                        else if (h2 == 3) a3 += p;
                        else              a4 += p;
                    }
                }
            }
        }
        out += swf[o * 5 + 0] * fmaxf(a0, 0.f)
             + swf[o * 5 + 1] * fmaxf(a1, 0.f)
             + swf[o * 5 + 2] * fmaxf(a2, 0.f)
             + swf[o * 5 + 3] * fmaxf(a3, 0.f)
             + swf[o * 5 + 4] * fmaxf(a4, 0.f);
    }
    score[(size_t)img * DD + d] = fmaxf(out, 0.f);
}

// -------------------------------------------------------------------------
// Attention + distance: one wave32 per (b,q,n) triple.
// Phase 1: acc[k] = sum_d tanh(Sf[k,d]*Qf[d])  (v_tanh_f32 + shuffle reduce)
// Phase 2: softmax over K=5 (redundant per lane), S_repre, weighted L2.
// -------------------------------------------------------------------------
__global__ void __launch_bounds__(256)
att_kernel(const float* __restrict__ S,     // B,N,K,D
           const float* __restrict__ Q,     // B,NQ,D
           const float* __restrict__ Sf,    // B,N,K,D
           const float* __restrict__ Qf,    // B,NQ,D
           const float* __restrict__ score, // B,N,D
           float* __restrict__ logits) {    // B,NQ,N
    const int wavesPerBlock = blockDim.x >> 5;
    const int w    = blockIdx.x * wavesPerBlock + (threadIdx.x >> 5);
    const int lane = threadIdx.x & 31;
    if (w >= BB * NQ * NN) return;

    const int b = w / (NQ * NN);
    const int r = w % (NQ * NN);
    const int q = r / NN;
    const int n = r % NN;

    const float* Sb  = S     + (size_t)(b * NN + n) * KK * DD;
    const float* Sfb = Sf    + (size_t)(b * NN + n) * KK * DD;
    const float* Qb  = Q     + (size_t)(b * NQ + q) * DD;
    const float* Qfb = Qf    + (size_t)(b * NQ + q) * DD;
    const float* sc  = score + (size_t)(b * NN + n) * DD;

    float acc0 = 0.f, acc1 = 0.f, acc2 = 0.f, acc3 = 0.f, acc4 = 0.f;
    for (int d = lane; d < DD; d += 32) {
        const float qf = Qfb[d];
        acc0 += dev_tanh(Sfb[0 * DD + d] * qf);
        acc1 += dev_tanh(Sfb[1 * DD + d] * qf);
        acc2 += dev_tanh(Sfb[2 * DD + d] * qf);
        acc3 += dev_tanh(Sfb[3 * DD + d] * qf);
        acc4 += dev_tanh(Sfb[4 * DD + d] * qf);
    }
#pragma unroll
    for (int off = 16; off > 0; off >>= 1) {
        acc0 += __shfl_xor(acc0, off, 32);
        acc1 += __shfl_xor(acc1, off, 32);
        acc2 += __shfl_xor(acc2, off, 32);
        acc3 += __shfl_xor(acc3, off, 32);
        acc4 += __shfl_xor(acc4, off, 32);
    }

    // softmax over K (all lanes have full sums)
    const float m  = fmaxf(fmaxf(fmaxf(acc0, acc1), fmaxf(acc2, acc3)), acc4);
    const float e0 = __expf(acc0 - m), e1 = __expf(acc1 - m),
                e2 = __expf(acc2 - m), e3 = __expf(acc3 - m),
                e4 = __expf(acc4 - m);
    const float inv = 1.f / (e0 + e1 + e2 + e3 + e4);
    const float a0 = e0 * inv, a1 = e1 * inv, a2 = e2 * inv,
                a3 = e3 * inv, a4 = e4 * inv;

    float part = 0.f;
    for (int d = lane; d < DD; d += 32) {
        const float sr = a0 * Sb[0 * DD + d] + a1 * Sb[1 * DD + d]
                       + a2 * Sb[2 * DD + d] + a3 * Sb[3 * DD + d]
                       + a4 * Sb[4 * DD + d];
        const float diff = sr - Qb[d];
        part += diff * diff * sc[d];
    }
#pragma unroll
    for (int off = 16; off > 0; off >>= 1) part += __shfl_xor(part, off, 32);

    if (lane == 0) logits[(size_t)(b * NQ + q) * NN + n] = -700.f * part;
}

// -------------------------------------------------------------------------
extern "C" void kernel_launch(void* const* d_in, const int* in_sizes, int n_in,
                              void* d_out, int out_size, void* d_ws, size_t ws_size,
                              hipStream_t stream) {
    const float* S    = (const float*)d_in[0];
    const float* Q    = (const float*)d_in[1];
    const float* fc_w = (const float*)d_in[2];
    const float* fc_b = (const float*)d_in[3];
    const float* w1   = (const float*)d_in[4];
    const float* b1   = (const float*)d_in[5];
    const float* w2   = (const float*)d_in[6];
    const float* b2   = (const float*)d_in[7];
    const float* wf   = (const float*)d_in[8];
    const float* bf   = (const float*)d_in[9];
    float* out = (float*)d_out;

    // workspace layout (floats): Qf[1600*230] | Sf[800*230] | score[160*230]
    float* Qf    = (float*)d_ws;
    float* Sf    = Qf + (size_t)BB * NQ * DD;
    float* score = Sf + (size_t)BB * NN * KK * DD;

    // Qf = Q @ fc_w^T + fc_b   (M = B*NQ = 1600)
    {
        dim3 grid((BB * NQ) / 16, (DD + 63) / 64);   // 100 x 4
        fc_wmma_kernel<<<grid, 32, 0, stream>>>(Q, fc_w, fc_b, Qf, BB * NQ);
    }
    // Sf = S @ fc_w^T + fc_b   (M = B*N*K = 800)
    {
        dim3 grid((BB * NN * KK) / 16, (DD + 63) / 64); // 50 x 4
        fc_wmma_kernel<<<grid, 32, 0, stream>>>(S, fc_w, fc_b, Sf, BB * NN * KK);
    }
    // score (B*N = 160 images, one block each)
    score_kernel<<<BB * NN, 256, 0, stream>>>(S, w1, b1, w2, b2, wf, bf, score);

    // attention + logits: one wave per (b,q,n)
    {
        const int waves = BB * NQ * NN;          // 32000
        att_kernel<<<(waves + 7) / 8, 256, 0, stream>>>(S, Q, Sf, Qf, score, out);
    }

    // second output: S passthrough
    hipMemcpyAsync(out + (size_t)BB * NQ * NN, S,
                   (size_t)BB * NN * KK * DD * sizeof(float),
                   hipMemcpyDeviceToDevice, stream);
}